// ExtremePatchMemory_35012573397051
// MI455X (gfx1250) — compile-verified
//
#include <hip/hip_runtime.h>
#include <math.h>

typedef __attribute__((ext_vector_type(2))) float v2f;
typedef __attribute__((ext_vector_type(8))) float v8f;

#define D_MODEL 256
#define NITEMS  16
#define KSTEPS  64               // 256 / 4 (K per wmma = 4)
#define TOPK_N  4
#define INV_TAU 10.0f
#define WAVES_PER_BLOCK 8
#define BLOCK_THREADS (WAVES_PER_BLOCK * 32)
#define ROWS_PER_BLOCK (WAVES_PER_BLOCK * 16)
#define MEM_PAD 264              // padded row stride (floats) for conflict-free gather

__global__ __launch_bounds__(BLOCK_THREADS)
void epm_topk_retrieve_kernel(const float* __restrict__ q,
                              const float* __restrict__ mem,
                              float* __restrict__ retrieved,
                              float* __restrict__ simmax)
{
    __shared__ float lds_mem[NITEMS][MEM_PAD];              // raw memory rows (retrieval)
    __shared__ float lds_minv[NITEMS];                      // 1/||mem_row||
    __shared__ v2f   lds_B[32 * (KSTEPS + 1)];              // normalized B fragments, padded
    __shared__ float lds_sim[WAVES_PER_BLOCK][NITEMS * 17]; // sim tiles, padded rows
    __shared__ float lds_w[WAVES_PER_BLOCK][NITEMS][TOPK_N];
    __shared__ int   lds_i[WAVES_PER_BLOCK][NITEMS][TOPK_N];

    const int tid  = threadIdx.x;
    const int wave = tid >> 5;
    const int lane = tid & 31;

    // ---- stage raw memory bank into LDS ----
    for (int t = tid; t < NITEMS * D_MODEL; t += BLOCK_THREADS) {
        int r = t >> 8, c = t & 255;
        lds_mem[r][c] = mem[t];
    }
    __syncthreads();

    // ---- per-row inverse norms of memory bank ----
    if (tid < NITEMS) {
        float s = 0.f;
        for (int c = 0; c < D_MODEL; ++c) { float x = lds_mem[tid][c]; s += x * x; }
        lds_minv[tid] = 1.0f / fmaxf(sqrtf(s), 1e-12f);
    }
    __syncthreads();

    // ---- build normalized B fragments in wmma lane layout ----
    // wmma lane s: row = s&15, K%4 in {0,1} for s<16, {2,3} for s>=16
    for (int t = tid; t < 32 * KSTEPS; t += BLOCK_THREADS) {
        int s  = t >> 6;
        int kk = t & 63;
        int row = s & 15;
        int c0 = 4 * kk + ((s & 16) ? 2 : 0);
        float inv = lds_minv[row];
        v2f b;
        b.x = lds_mem[row][c0]     * inv;
        b.y = lds_mem[row][c0 + 1] * inv;
        lds_B[s * (KSTEPS + 1) + kk] = b;
    }
    __syncthreads();

    // ---- per-wave 16-row tile: sim = (q_norm . m_norm) / tau via f32 WMMA ----
    const int row0  = blockIdx.x * ROWS_PER_BLOCK + wave * 16;
    const int myrow = row0 + (lane & 15);
    const int c0    = (lane & 16) ? 2 : 0;
    const float* qp = q + (size_t)myrow * D_MODEL + c0;

    v8f acc = {};
    float sumsq = 0.f;
#pragma unroll 8
    for (int kk = 0; kk < KSTEPS; ++kk) {
        v2f a = *(const v2f*)(qp + 4 * kk);
        sumsq += a.x * a.x + a.y * a.y;
        v2f b = lds_B[lane * (KSTEPS + 1) + kk];
        // D = A(16x4 f32) * B(4x16 f32) + C(16x16 f32)
        acc = __builtin_amdgcn_wmma_f32_16x16x4_f32(
            /*neg_a=*/false, a, /*neg_b=*/false, b,
            /*c_mod=*/(short)0, acc, /*reuse_a=*/false, /*reuse_b=*/false);
    }

    // combine the two half-row partial sums of |q|^2, build per-row scale
    sumsq += __shfl_xor(sumsq, 16);
    float scale = INV_TAU / fmaxf(sqrtf(sumsq), 1e-12f);

    // scale accumulator rows and spill sim tile to LDS
    const int hi = (lane & 16) ? 8 : 0;    // rows held by this half: hi + v
    float* simb = lds_sim[wave];
#pragma unroll
    for (int v = 0; v < 8; ++v) {
        float s = __shfl(scale, v + hi);   // row (v+hi)'s scale lives in lane (v+hi)
        simb[(v + hi) * 17 + (lane & 15)] = acc[v] * s;
    }
    __syncthreads();

    // ---- top-4 + softmax + sim_max, one row per low lane ----
    if (lane < 16) {
        const int r = lane;
        float vals[16];
#pragma unroll
        for (int i = 0; i < 16; ++i) vals[i] = simb[r * 17 + i];
        float tv[TOPK_N]; int ti[TOPK_N];
#pragma unroll
        for (int k = 0; k < TOPK_N; ++k) {
            float bv = -__builtin_inff(); int bi = 0;
#pragma unroll
            for (int i = 0; i < 16; ++i) {
                if (vals[i] > bv) { bv = vals[i]; bi = i; }
            }
            tv[k] = bv; ti[k] = bi; vals[bi] = -__builtin_inff();
        }
        simmax[row0 + r] = tv[0];
        float mx = tv[0];
        float e[TOPK_N]; float es = 0.f;
#pragma unroll
        for (int k = 0; k < TOPK_N; ++k) { e[k] = expf(tv[k] - mx); es += e[k]; }
        float inv = 1.0f / es;
#pragma unroll
        for (int k = 0; k < TOPK_N; ++k) {
            lds_w[wave][r][k] = e[k] * inv;
            lds_i[wave][r][k] = ti[k];
        }
    }
    __syncthreads();

    // ---- retrieval: weighted gather of raw memory rows, coalesced stores ----
    for (int r = 0; r < 16; ++r) {
        float w0 = lds_w[wave][r][0], w1 = lds_w[wave][r][1];
        float w2 = lds_w[wave][r][2], w3 = lds_w[wave][r][3];
        int   i0 = lds_i[wave][r][0], i1 = lds_i[wave][r][1];
        int   i2 = lds_i[wave][r][2], i3 = lds_i[wave][r][3];
        float* op = retrieved + (size_t)(row0 + r) * D_MODEL;
#pragma unroll
        for (int d = lane; d < D_MODEL; d += 32) {
            op[d] = w0 * lds_mem[i0][d] + w1 * lds_mem[i1][d]
                  + w2 * lds_mem[i2][d] + w3 * lds_mem[i3][d];
        }
    }
}

extern "C" void kernel_launch(void* const* d_in, const int* in_sizes, int n_in,
                              void* d_out, int out_size, void* d_ws, size_t ws_size,
                              hipStream_t stream) {
    const float* q   = (const float*)d_in[0];   // [32,4096,256] f32
    const float* mem = (const float*)d_in[1];   // [16,256] f32 (unit rows)
    const int n_rows = in_sizes[0] / D_MODEL;   // 131072
    float* retrieved = (float*)d_out;                              // [N,256]
    float* simmax    = (float*)d_out + (size_t)n_rows * D_MODEL;   // [N,1]
    const int grid = n_rows / ROWS_PER_BLOCK;                      // 1024
    epm_topk_retrieve_kernel<<<grid, BLOCK_THREADS, 0, stream>>>(q, mem, retrieved, simmax);
}